// PointBackbone_57913339019735
// MI455X (gfx1250) — compile-verified
//
#include <hip/hip_runtime.h>
#include <hip/hip_bf16.h>

// ---------------------------------------------------------------------------
// KPConv backbone for MI455X (gfx1250, wave32).
// Dense contractions use V_WMMA_F32_16X16X4_F32 (full f32 on the matrix pipe).
// GEMM: branch-free, software-pipelined, 32-bit offsets off uniform bases so
// loads lower to global_load (saddr + voffset), not flat_load.
// ---------------------------------------------------------------------------

typedef float v2f __attribute__((ext_vector_type(2)));
typedef float v8f __attribute__((ext_vector_type(8)));

#define KNBR 32
#define NKP 15
#define LRELU(v) ((v) >= 0.f ? (v) : 0.1f * (v))

// ---------------------------------------------------------------------------
// GEMM: C[M,N] = A[M,Kd] @ B[Kd,N], optional bias[N], optional rowscale[M].
// Kd % 4 == 0 (both operands zero-padded in K). One wave computes a
// (16*MT) x (16*NT) strip; B fetches are shared across the MT row-tiles.
// All element offsets < 2^31 -> 32-bit offset arithmetic, uniform SGPR bases.
// A (16x4 f32): lane row = lane&15, VGPR v holds K = kk + v + 2*(lane>>4)
// B (4x16 f32): lane col = lane&15, VGPR v holds row K = kk + v + 2*(lane>>4)
// D (16x16 f32): VGPR v holds row M = v + 8*(lane>>4), col = lane&15
// ---------------------------------------------------------------------------
template <int MT, int NT>
__global__ __launch_bounds__(32) void k_gemm16(
    const float* __restrict__ A, const float* __restrict__ B,
    const float* __restrict__ bias, const float* __restrict__ rowscale,
    float* __restrict__ Cmat, int M, int N, int Kd)
{
    const int m0    = blockIdx.x * (16 * MT);
    const int n0    = blockIdx.y * (16 * NT);
    const int lane  = threadIdx.x & 31;
    const int half  = lane >> 4;
    const int lid   = lane & 15;
    const int kbase = half * 2;

    // 32-bit element offsets (edge tiles clamped once, outside the loop)
    int offA[MT];
    #pragma unroll
    for (int s = 0; s < MT; ++s) {
        const int mr = m0 + s * 16 + lid;
        const int mc = mr < M ? mr : (M - 1);
        offA[s] = mc * Kd + kbase;
    }

    int ncol[NT], offB0[NT], offB1[NT];
    #pragma unroll
    for (int t = 0; t < NT; ++t) {
        ncol[t] = n0 + t * 16 + lid;
        const int nc = ncol[t] < N ? ncol[t] : (N - 1);
        offB0[t] = kbase * N + nc;
        offB1[t] = offB0[t] + N;
    }

    v8f acc[MT][NT];
    #pragma unroll
    for (int s = 0; s < MT; ++s)
        #pragma unroll
        for (int t = 0; t < NT; ++t)
            acc[s][t] = (v8f){0.f,0.f,0.f,0.f,0.f,0.f,0.f,0.f};

    const int dB = 4 * N;

    // ---- preload first K-block ----
    v2f   a[MT];
    float b0[NT], b1[NT];
    #pragma unroll
    for (int s = 0; s < MT; ++s) a[s] = *(const v2f*)(A + offA[s]);
    #pragma unroll
    for (int t = 0; t < NT; ++t) { b0[t] = B[offB0[t]]; b1[t] = B[offB1[t]]; }

    const int nb = Kd >> 2;
    #pragma unroll 2
    for (int it = 0; it < nb - 1; ++it) {
        // issue loads for next K-block (32-bit offset increments only)
        v2f   an[MT];
        float nb0[NT], nb1[NT];
        #pragma unroll
        for (int s = 0; s < MT; ++s) { offA[s] += 4; an[s] = *(const v2f*)(A + offA[s]); }
        #pragma unroll
        for (int t = 0; t < NT; ++t) {
            offB0[t] += dB; offB1[t] += dB;
            nb0[t] = B[offB0[t]]; nb1[t] = B[offB1[t]];
        }
        // WMMA on current K-block (MT*NT back-to-back, distinct accumulators)
        #pragma unroll
        for (int t = 0; t < NT; ++t) {
            v2f bv; bv.x = b0[t]; bv.y = b1[t];
            #pragma unroll
            for (int s = 0; s < MT; ++s)
                acc[s][t] = __builtin_amdgcn_wmma_f32_16x16x4_f32(
                    false, a[s], false, bv, (short)0, acc[s][t], false, false);
        }
        #pragma unroll
        for (int s = 0; s < MT; ++s) a[s] = an[s];
        #pragma unroll
        for (int t = 0; t < NT; ++t) { b0[t] = nb0[t]; b1[t] = nb1[t]; }
    }
    // tail block (no prefetch)
    #pragma unroll
    for (int t = 0; t < NT; ++t) {
        v2f bv; bv.x = b0[t]; bv.y = b1[t];
        #pragma unroll
        for (int s = 0; s < MT; ++s)
            acc[s][t] = __builtin_amdgcn_wmma_f32_16x16x4_f32(
                false, a[s], false, bv, (short)0, acc[s][t], false, false);
    }

    // ---- epilogue ----
    float rsv[MT][8];
    #pragma unroll
    for (int s = 0; s < MT; ++s)
        #pragma unroll
        for (int v = 0; v < 8; ++v) {
            const int m = m0 + s * 16 + v + half * 8;
            rsv[s][v] = (rowscale && m < M) ? rowscale[m] : 1.f;
        }
    #pragma unroll
    for (int t = 0; t < NT; ++t) {
        if (ncol[t] < N) {
            const float bval = bias ? bias[ncol[t]] : 0.f;
            #pragma unroll
            for (int s = 0; s < MT; ++s) {
                #pragma unroll
                for (int v = 0; v < 8; ++v) {
                    const int m = m0 + s * 16 + v + half * 8;
                    if (m < M)
                        Cmat[m * N + ncol[t]] = acc[s][t][v] * rsv[s][v] + bval;
                }
            }
        }
    }
}

// zero-pad a row-major [rows,cols] matrix to [rowsPad,cols]
__global__ void k_padrows(const float* __restrict__ src, float* __restrict__ dst,
                          int nsrc, int ndst)
{
    const int i = blockIdx.x * blockDim.x + threadIdx.x;
    if (i < ndst) dst[i] = (i < nsrc) ? src[i] : 0.f;
}

// ---------------------------------------------------------------------------
// KPConv gather: weighted[m, p, c] = sum_k infl(m,k,p) * feat(nbr(m,k), c)
// C is a power of two (1..256): div/mod by C are shift/mask (Cl2 = log2 C).
// K dim padded to PCp (multiple of 4), padding written as zeros.
// ---------------------------------------------------------------------------
__global__ __launch_bounds__(128) void k_kpconv_weighted(
    const float* __restrict__ qp, const float* __restrict__ sp,
    const float* __restrict__ sf, const int* __restrict__ nbr,
    const float* __restrict__ kpts,
    float* __restrict__ wout, float* __restrict__ invn,
    int M, int Ns, int C, int Cl2, int PCp, float sigma)
{
    const int m   = blockIdx.x;
    const int tid = threadIdx.x;

    __shared__ float s_feat[256];
    __shared__ float s_infl[NKP];
    __shared__ float s_red[128];
    __shared__ float s_kp[NKP * 3];
    __shared__ float s_q[3];
    __shared__ int   s_cnt;

    if (tid < NKP * 3) s_kp[tid] = kpts[tid];
    if (tid < 3)       s_q[tid]  = qp[m * 3 + tid];
    if (tid == 0)      s_cnt = 0;

    const int PC    = NKP * C;
    const int per   = (PCp + 127) >> 7;   // <= 30 (C <= 256)
    const int cmask = C - 1;
    float acc[30];
    for (int i = 0; i < per; ++i) acc[i] = 0.f;
    __syncthreads();

    for (int k = 0; k < KNBR; ++k) {
        const int  idx   = nbr[m * KNBR + k];
        const bool valid = (idx >= 0) && (idx < Ns);

        for (int c = tid; c < C; c += 128)
            s_feat[c] = valid ? sf[idx * C + c] : 0.f;

        if (tid < NKP) {
            float infl = 0.f;
            if (valid) {
                const float rx = sp[idx * 3 + 0] - s_q[0];
                const float ry = sp[idx * 3 + 1] - s_q[1];
                const float rz = sp[idx * 3 + 2] - s_q[2];
                const float dx = rx - s_kp[tid * 3 + 0];
                const float dy = ry - s_kp[tid * 3 + 1];
                const float dz = rz - s_kp[tid * 3 + 2];
                const float d  = sqrtf(dx * dx + dy * dy + dz * dz);
                infl = fmaxf(1.f - d / sigma, 0.f);
            }
            s_infl[tid] = infl;
        }
        __syncthreads();

        // neighbor feature-sum reduction (for nnum)
        float ps = 0.f;
        for (int c = tid; c < C; c += 128) ps += s_feat[c];
        s_red[tid] = ps;
        __syncthreads();
        for (int t = 64; t > 0; t >>= 1) {
            if (tid < t) s_red[tid] += s_red[tid + t];
            __syncthreads();
        }
        if (tid == 0 && s_red[0] > 0.f) s_cnt++;

        for (int i = 0; i < per; ++i) {
            const int j = tid + (i << 7);
            if (j < PC) acc[i] += s_infl[j >> Cl2] * s_feat[j & cmask];
        }
        __syncthreads();
    }

    for (int i = 0; i < per; ++i) {
        const int j = tid + (i << 7);
        if (j < PCp) wout[m * PCp + j] = (j < PC) ? acc[i] : 0.f;
    }
    if (tid == 0) {
        const int cnt = s_cnt > 1 ? s_cnt : 1;
        invn[m] = 1.f / (float)cnt;
    }
}

// ---------------------------------------------------------------------------
// GroupNorm with GLOBAL statistics (mean/var over all rows, per group of 32).
// ---------------------------------------------------------------------------
__global__ __launch_bounds__(256) void k_gn_stats(
    const float* __restrict__ x, float* __restrict__ stats,
    int N, int C, float eps)
{
    __shared__ float rs[256], rq[256];
    const int  g     = blockIdx.x;
    const int  cg    = C >> 5;                 // power of two
    const int  csh   = __builtin_ctz(cg);
    const int  cmask = cg - 1;
    const int  total = N * cg;

    float s = 0.f, q = 0.f;
    for (int i = threadIdx.x; i < total; i += 256) {
        const int n = i >> csh;
        const int c = i & cmask;
        const float v = x[n * C + g * cg + c];
        s += v; q += v * v;
    }
    rs[threadIdx.x] = s; rq[threadIdx.x] = q;
    __syncthreads();
    for (int t = 128; t > 0; t >>= 1) {
        if (threadIdx.x < t) { rs[threadIdx.x] += rs[threadIdx.x + t];
                               rq[threadIdx.x] += rq[threadIdx.x + t]; }
        __syncthreads();
    }
    if (threadIdx.x == 0) {
        const float mean = rs[0] / (float)total;
        const float var  = rq[0] / (float)total - mean * mean;
        stats[g]      = mean;
        stats[32 + g] = rsqrtf(fmaxf(var, 0.f) + eps);
    }
}

__global__ void k_gn_apply(
    const float* __restrict__ x, float* __restrict__ y,
    const float* __restrict__ stats, const float* __restrict__ gamma,
    const float* __restrict__ beta, int N, int C, int relu)
{
    const int i = blockIdx.x * blockDim.x + threadIdx.x;
    if (i >= N * C) return;
    const int c   = i % C;                    // C is pow2 -> compiler folds
    const int csh = __builtin_ctz(C >> 5);
    const int g   = c >> csh;
    float v = (x[i] - stats[g]) * stats[32 + g] * gamma[c] + beta[c];
    if (relu) v = LRELU(v);
    y[i] = v;
}

__global__ void k_add_leaky(const float* __restrict__ a,
                            const float* __restrict__ b,
                            float* __restrict__ o, int n)
{
    const int i = blockIdx.x * blockDim.x + threadIdx.x;
    if (i < n) { const float v = a[i] + b[i]; o[i] = LRELU(v); }
}

__global__ void k_maxpool(const float* __restrict__ f, const int* __restrict__ nbr,
                          float* __restrict__ out, int M, int Ns, int C)
{
    const int i = blockIdx.x * blockDim.x + threadIdx.x;
    if (i >= M * C) return;
    const int m = i / C, c = i % C;
    float best = -3.402823e38f;
    for (int k = 0; k < KNBR; ++k) {
        const int idx = nbr[m * KNBR + k];
        const float v = (idx >= 0 && idx < Ns) ? f[idx * C + c] : 0.f;
        best = fmaxf(best, v);
    }
    out[i] = best;
}

__global__ void k_knn_interp(const float* __restrict__ qp, const float* __restrict__ sp,
                             const float* __restrict__ sf, const int* __restrict__ ups,
                             float* __restrict__ out, int M, int Ns, int C)
{
    const int i = blockIdx.x * blockDim.x + threadIdx.x;
    if (i >= M * C) return;
    const int m = i / C, c = i % C;
    const float qx = qp[m * 3 + 0], qy = qp[m * 3 + 1], qz = qp[m * 3 + 2];
    float w[3]; int id[3]; float ws = 0.f;
    #pragma unroll
    for (int j = 0; j < 3; ++j) {
        const int idx = ups[m * 3 + j];
        id[j] = idx;
        float sqd;
        if (idx >= 0 && idx < Ns) {
            const float dx = qx - sp[idx * 3 + 0];
            const float dy = qy - sp[idx * 3 + 1];
            const float dz = qz - sp[idx * 3 + 2];
            sqd = dx * dx + dy * dy + dz * dz;
        } else {
            sqd = 1e20f;
        }
        w[j] = 1.f / (sqd + 1e-8f);
        ws += w[j];
    }
    const float inv = 1.f / ws;
    float v = 0.f;
    #pragma unroll
    for (int j = 0; j < 3; ++j)
        if (id[j] >= 0 && id[j] < Ns) v += w[j] * inv * sf[id[j] * C + c];
    out[i] = v;
}

__global__ void k_concat(const float* __restrict__ A, const float* __restrict__ B,
                         float* __restrict__ out, int N, int ca, int cb)
{
    const int ct = ca + cb;
    const int i  = blockIdx.x * blockDim.x + threadIdx.x;
    if (i >= N * ct) return;
    const int n = i / ct, c = i % ct;
    out[i] = (c < ca) ? A[n * ca + c] : B[n * cb + (c - ca)];
}

// ---------------------------------------------------------------------------
// Host-side orchestration
// ---------------------------------------------------------------------------
namespace {

struct Lin  { const float *b, *be, *g, *w; };
struct ConvP{ const float *be, *g, *kpts, *kw; };
struct ResP { const float *kpts, *kw, *nbe, *ng; Lin sc; bool has_sc; Lin u1, u2; };

inline const float* nf(void* const* d, int& i) { return (const float*)d[i++]; }
inline Lin rdLin(void* const* d, int& i) {
    Lin l; l.b = nf(d,i); l.be = nf(d,i); l.g = nf(d,i); l.w = nf(d,i); return l;
}
inline ConvP rdConv(void* const* d, int& i) {
    ConvP c; c.be = nf(d,i); c.g = nf(d,i); c.kpts = nf(d,i); c.kw = nf(d,i); return c;
}
inline ResP rdRes(void* const* d, int& i, bool sc) {
    ResP r; r.kpts = nf(d,i); r.kw = nf(d,i); r.nbe = nf(d,i); r.ng = nf(d,i);
    r.has_sc = sc;
    if (sc) r.sc = rdLin(d,i);
    r.u1 = rdLin(d,i); r.u2 = rdLin(d,i);
    return r;
}

struct Ws {
    float *bufW;   // weighted [<=40000 x 480]; reused as interp buffer in decoder
    float *bufC;   // concat buffer [<=40000 x 384]; xc/scA/scB alias into it (encoder)
    float *xa, *xb, *xc, *scA, *scB;
    float *invn, *stats, *kwpad;
};

// Kd must be a multiple of 4 on both operands.
inline void gemm(const float* A, const float* B, const float* bias, const float* rs,
                 float* C, int M, int N, int Kd, hipStream_t st)
{
    const int mt = (M + 31) / 32;
    if (N >= 64) {
        dim3 g(mt, (N + 63) / 64);
        k_gemm16<2,4><<<g, 32, 0, st>>>(A, B, bias, rs, C, M, N, Kd);
    } else if (N >= 32) {
        dim3 g(mt, (N + 31) / 32);
        k_gemm16<2,2><<<g, 32, 0, st>>>(A, B, bias, rs, C, M, N, Kd);
    } else {
        dim3 g(mt, (N + 15) / 16);
        k_gemm16<2,1><<<g, 32, 0, st>>>(A, B, bias, rs, C, M, N, Kd);
    }
}

inline void groupnorm(float* x, int N, int C, const float* gamma, const float* beta,
                      int relu, float* stats, hipStream_t st)
{
    k_gn_stats<<<32, 256, 0, st>>>(x, stats, N, C, 1e-5f);
    const int tot = N * C;
    k_gn_apply<<<(tot + 255) / 256, 256, 0, st>>>(x, x, stats, gamma, beta, N, C, relu);
}

inline void unary(const float* x, float* y, int N, int Cin, int Cout,
                  const Lin& p, int relu, float* stats, hipStream_t st)
{
    gemm(x, p.w, p.b, nullptr, y, N, Cout, Cin, st);
    groupnorm(y, N, Cout, p.g, p.be, relu, stats, st);
}

inline void kpconv(const float* qp, const float* sp, const float* sf, const int* nbr,
                   const float* kpts, const float* kw, int M, int Ns, int Cin, int Cout,
                   float sigma, Ws& w, float* out, hipStream_t st)
{
    const int PC  = NKP * Cin;
    const int PCp = (PC + 3) & ~3;
    const int cl2 = __builtin_ctz(Cin);       // Cin is a power of two
    k_kpconv_weighted<<<M, 128, 0, st>>>(qp, sp, sf, nbr, kpts, w.bufW, w.invn,
                                         M, Ns, Cin, cl2, PCp, sigma);
    const float* Bmat = kw;
    if (PCp != PC) {  // only e1_1 (PC=15): zero-pad kw rows to PCp
        const int nsrc = PC * Cout, ndst = PCp * Cout;
        k_padrows<<<(ndst + 255) / 256, 256, 0, st>>>(kw, w.kwpad, nsrc, ndst);
        Bmat = w.kwpad;
    }
    gemm(w.bufW, Bmat, nullptr, w.invn, out, M, Cout, PCp, st);
}

inline void resblock(const float* q, const float* s, const float* fin, const int* nbr,
                     const ResP& p, int Nq, int Ns, int Cin, int Cout, float sigma,
                     bool strided, float* out, Ws& w, hipStream_t st)
{
    const int mid = Cout / 4;
    // u1: [Ns,Cin] -> [Ns,mid]
    unary(fin, w.xa, Ns, Cin, mid, p.u1, 1, w.stats, st);
    // kpconv: -> [Nq,mid], then GN+leaky
    kpconv(q, s, w.xa, nbr, p.kpts, p.kw, Nq, Ns, mid, mid, sigma, w, w.xb, st);
    groupnorm(w.xb, Nq, mid, p.ng, p.nbe, 1, w.stats, st);
    // u2 (no relu): -> [Nq,Cout]
    unary(w.xb, w.xc, Nq, mid, Cout, p.u2, 0, w.stats, st);
    // shortcut
    const float* sc = fin;
    if (strided) {
        const int tot = Nq * Cin;
        k_maxpool<<<(tot + 255) / 256, 256, 0, st>>>(fin, nbr, w.scA, Nq, Ns, Cin);
        sc = w.scA;
    }
    if (p.has_sc) {
        unary(sc, w.scB, Nq, Cin, Cout, p.sc, 0, w.stats, st);
        sc = w.scB;
    }
    const int tot = Nq * Cout;
    k_add_leaky<<<(tot + 255) / 256, 256, 0, st>>>(w.xc, sc, out, tot);
}

} // namespace

extern "C" void kernel_launch(void* const* d_in, const int* in_sizes, int n_in,
                              void* d_out, int out_size, void* d_ws, size_t ws_size,
                              hipStream_t stream)
{
    (void)in_sizes; (void)n_in; (void)out_size; (void)ws_size;
    const int N0 = 40000, N1 = 10000, N2 = 2500, N3 = 640;
    const float S = 0.025f * 2.0f;   // SIGMA

    // ---- inputs (jax-tree order: dict keys sorted recursively inside params) ----
    int ii = 0;
    const float* feats = nf(d_in, ii);
    Lin   d1   = rdLin(d_in, ii);
    Lin   d2   = rdLin(d_in, ii);
    Lin   d3   = rdLin(d_in, ii);
    ConvP e1_1 = rdConv(d_in, ii);
    ResP  e1_2 = rdRes(d_in, ii, true);
    ResP  e2_1 = rdRes(d_in, ii, false);
    ResP  e2_2 = rdRes(d_in, ii, true);
    ResP  e2_3 = rdRes(d_in, ii, false);
    ResP  e3_1 = rdRes(d_in, ii, false);
    ResP  e3_2 = rdRes(d_in, ii, true);
    ResP  e3_3 = rdRes(d_in, ii, false);
    ResP  e4_1 = rdRes(d_in, ii, false);
    ResP  e4_2 = rdRes(d_in, ii, true);
    ResP  e4_3 = rdRes(d_in, ii, false);
    const float* out_b = nf(d_in, ii);     // params['out']['b']
    const float* out_w = nf(d_in, ii);     // params['out']['w']

    const float* pts0 = nf(d_in, ii); const int* nbr0 = (const int*)d_in[ii++];
    const float* pts1 = nf(d_in, ii); const int* nbr1 = (const int*)d_in[ii++];
    const float* pts2 = nf(d_in, ii); const int* nbr2 = (const int*)d_in[ii++];
    const float* pts3 = nf(d_in, ii); const int* nbr3 = (const int*)d_in[ii++];
    const int* sub0 = (const int*)d_in[ii++]; const int* up0 = (const int*)d_in[ii++];
    const int* sub1 = (const int*)d_in[ii++]; const int* up1 = (const int*)d_in[ii++];
    const int* sub2 = (const int*)d_in[ii++]; const int* up2 = (const int*)d_in[ii++];

    // ---- outputs (l1, l2, l3, l4 flattened in return order) ----
    float* l1 = (float*)d_out;
    float* l2 = l1 + (size_t)N0 * 256;
    float* l3 = l2 + (size_t)N1 * 256;
    float* l4 = l3 + (size_t)N2 * 512;

    // ---- workspace layout (bump allocator) ----
    char* base = (char*)d_ws;
    size_t off = 0;
    auto alloc = [&](size_t nfloats) -> float* {
        float* p = (float*)(base + off);
        off += ((nfloats * sizeof(float) + 255) / 256) * 256;
        return p;
    };
    Ws w;
    w.bufW  = alloc((size_t)N0 * NKP * 32);   // largest weighted buffer
    w.bufC  = alloc((size_t)N0 * 384);        // decoder concat buffer
    float* g1   = alloc((size_t)N0 * 128);    // f1 (kept for decoder)
    float* tmp1 = alloc((size_t)N0 * 128);    // conv out / d1 out
    float* B2a  = alloc((size_t)N1 * 256);
    float* B2b  = alloc((size_t)N1 * 256);
    float* B3a  = alloc((size_t)N2 * 512);
    float* B3b  = alloc((size_t)N2 * 512);
    float* B4a  = alloc((size_t)N3 * 1024);
    float* B4b  = alloc((size_t)N3 * 1024);
    w.xa    = alloc((size_t)N0 * 32);
    w.xb    = alloc((size_t)N0 * 32);
    w.invn  = alloc((size_t)N0);
    w.stats = alloc(64);
    w.kwpad = alloc(16 * 64);                 // padded e1_1 kernel weights
    // encoder-only temps alias into bufC (decoder-only buffer)
    w.xc  = w.bufC;                                          // <= N0*128
    w.scA = w.bufC + (size_t)N0 * 128;                       // <= N1*128
    w.scB = w.bufC + (size_t)N0 * 128 + (size_t)N1 * 128;    // <= N0*128

    // ================= Encoder =================
    // e1_1: conv block, Cin=1 -> 64
    kpconv(pts0, pts0, feats, nbr0, e1_1.kpts, e1_1.kw, N0, N0, 1, 64, S, w, tmp1, stream);
    groupnorm(tmp1, N0, 64, e1_1.g, e1_1.be, 1, w.stats, stream);
    // e1_2: 64 -> 128
    resblock(pts0, pts0, tmp1, nbr0, e1_2, N0, N0, 64, 128, S, false, g1, w, stream);

    // level 2
    resblock(pts1, pts0, g1,  sub0, e2_1, N1, N0, 128, 128, S,       true,  B2a, w, stream);
    resblock(pts1, pts1, B2a, nbr1, e2_2, N1, N1, 128, 256, 2.f * S, false, B2b, w, stream);
    resblock(pts1, pts1, B2b, nbr1, e2_3, N1, N1, 256, 256, 2.f * S, false, B2a, w, stream);
    // level 3
    resblock(pts2, pts1, B2a, sub1, e3_1, N2, N1, 256, 256, 2.f * S, true,  B3a, w, stream);
    resblock(pts2, pts2, B3a, nbr2, e3_2, N2, N2, 256, 512, 4.f * S, false, B3b, w, stream);
    resblock(pts2, pts2, B3b, nbr2, e3_3, N2, N2, 512, 512, 4.f * S, false, B3a, w, stream);
    // level 4
    resblock(pts3, pts2, B3a, sub2, e4_1, N3, N2, 512, 512,  4.f * S, true,  B4a, w, stream);
    resblock(pts3, pts3, B4a, nbr3, e4_2, N3, N3, 512, 1024, 8.f * S, false, B4b, w, stream);
    resblock(pts3, pts3, B4b, nbr3, e4_3, N3, N3, 1024, 1024, 8.f * S, false, l4, w, stream);
    // f1 = g1, f2 = B2a, f3 = B3a, f4 = l4

    // ================= Decoder =================
    float* interp = w.bufW;   // bufW free after encoder; >= N0*256 floats

    // l3 = unary(concat(knn(pts2<-pts3, l4), f3), d3)
    {
        int t = N2 * 1024;
        k_knn_interp<<<(t + 255) / 256, 256, 0, stream>>>(pts2, pts3, l4, up2,
                                                          interp, N2, N3, 1024);
        t = N2 * 1536;
        k_concat<<<(t + 255) / 256, 256, 0, stream>>>(interp, B3a, w.bufC, N2, 1024, 512);
        unary(w.bufC, l3, N2, 1536, 512, d3, 1, w.stats, stream);
    }
    // l2 = unary(concat(knn(pts1<-pts2, l3), f2), d2)
    {
        int t = N1 * 512;
        k_knn_interp<<<(t + 255) / 256, 256, 0, stream>>>(pts1, pts2, l3, up1,
                                                          interp, N1, N2, 512);
        t = N1 * 768;
        k_concat<<<(t + 255) / 256, 256, 0, stream>>>(interp, B2a, w.bufC, N1, 512, 256);
        unary(w.bufC, l2, N1, 768, 256, d2, 1, w.stats, stream);
    }
    // l1 = unary(concat(knn(pts0<-pts1, l2), f1), d1) @ out_w + out_b
    {
        int t = N0 * 256;
        k_knn_interp<<<(t + 255) / 256, 256, 0, stream>>>(pts0, pts1, l2, up0,
                                                          interp, N0, N1, 256);
        t = N0 * 384;
        k_concat<<<(t + 255) / 256, 256, 0, stream>>>(interp, g1, w.bufC, N0, 256, 128);
        unary(w.bufC, tmp1, N0, 384, 128, d1, 1, w.stats, stream);
        gemm(tmp1, out_w, out_b, nullptr, l1, N0, 256, 128, stream);
    }
}